// CapsuleLayer_79929341379029
// MI455X (gfx1250) — compile-verified
//
#include <hip/hip_runtime.h>
#include <hip/hip_bf16.h>
#include <math.h>

typedef float v2f  __attribute__((ext_vector_type(2)));
typedef float v4f  __attribute__((ext_vector_type(4)));
typedef float v8f  __attribute__((ext_vector_type(8)));

// Problem constants
#define Bsz 32
#define Nsz 2048
#define Isz 64
#define Ksz 32
#define Dsz 64
#define NCHUNK 32              // n-chunks for routing partials
#define NPERCHUNK (Nsz / NCHUNK)
#define GCHUNK 64              // GEMM n-chunks (32 n each)
#define GCH2   (GCHUNK * 2)    // partial slots (per n-parity)

// ---------------------------------------------------------------------------
// Phase 1 (+ fused routing iter-0 partial sums):
//   u_hat[k][n][b][d] = sum_i x[b][n][i] * W[k][n][d][i]
//   spart1[k][ch][b][d] = sum_{n in slot ch} u_hat[k][n][b][d]
// Block = (k, 32-n chunk). 8 waves = 2 n in flight x 4 d-tiles.
// f32 WMMA 16x16x4, K-loop of 16 steps over I=64.
// ---------------------------------------------------------------------------
__global__ __launch_bounds__(256) void uhat_gemm(const float* __restrict__ x,
                                                 const float* __restrict__ W,
                                                 float* __restrict__ uh,
                                                 float* __restrict__ spart1) {
  __shared__ float xs[2 * Bsz * Isz];   // 16 KB: two x[:, n, :] slices
  const int k     = blockIdx.x >> 6;    // 32
  const int nc    = blockIdx.x & 63;    // 64 chunks of 32 n
  const int nbase = nc * 32;
  const int t     = threadIdx.x;
  const int wave  = t >> 5;
  const int lane  = t & 31;
  const int l15   = lane & 15;
  const int lhi   = lane >> 4;          // 0 or 1
  const int npar  = wave >> 2;          // which of the 2 in-flight n
  const int d0    = (wave & 3) << 4;    // d-tile

  v8f sacc0 = {};
  v8f sacc1 = {};

  const float* xp0 = xs + npar * (Bsz * Isz) + l15 * Isz;  // b-tile 0
  const float* xp1 = xp0 + 16 * Isz;                       // b-tile 1
  const int drow = d0 + (lhi << 3);

  for (int nn = 0; nn < 32; nn += 2) {
    __syncthreads();   // xs reuse
    { // stage x[:, n0, :] and x[:, n0+1, :]
      const int b  = t >> 3;
      const int i0 = (t & 7) * 8;
      const float* s0 = x + ((size_t)b * Nsz + nbase + nn) * Isz + i0;
      v4f a0 = *(const v4f*)(s0);
      v4f a1 = *(const v4f*)(s0 + 4);
      v4f c0 = *(const v4f*)(s0 + Isz);
      v4f c1 = *(const v4f*)(s0 + Isz + 4);
      *(v4f*)(xs + b * Isz + i0)                 = a0;
      *(v4f*)(xs + b * Isz + i0 + 4)             = a1;
      *(v4f*)(xs + Bsz * Isz + b * Isz + i0)     = c0;
      *(v4f*)(xs + Bsz * Isz + b * Isz + i0 + 4) = c1;
    }
    __syncthreads();

    const int n = nbase + nn + npar;
    v8f acc0 = {};
    v8f acc1 = {};
    // A fragment base: row (d0+l15) of W[k][n], col offset 2*lhi (ISA 16x4 layout)
    const float* wp = W + (((size_t)k * Nsz + n) * Dsz + d0 + l15) * Isz + (lhi << 1);

#pragma unroll
    for (int s = 0; s < 16; ++s) {
      const int i0 = s << 2;
      v2f A  = *(const v2f*)(wp + i0);
      const int r0 = i0 + (lhi << 1);        // B rows {0,1}/{2,3} per half-wave
      v2f B0 = *(const v2f*)(xp0 + r0);
      v2f B1 = *(const v2f*)(xp1 + r0);
      acc0 = __builtin_amdgcn_wmma_f32_16x16x4_f32(false, A, false, B0,
                                                   (short)0, acc0, false, false);
      acc1 = __builtin_amdgcn_wmma_f32_16x16x4_f32(false, A, false, B1,
                                                   (short)0, acc1, false, false);
    }

    // store u_hat tile (streaming -> non-temporal), C/D layout:
    // lane col = l15 (=b'), rows j + 8*lhi (=d')
    float* out = uh + ((size_t)k * Nsz + n) * (Bsz * Dsz);
    float* o0 = out + l15 * Dsz + drow;
    float* o1 = out + (16 + l15) * Dsz + drow;
    __builtin_nontemporal_store(((v4f*)&acc0)[0], (v4f*)(o0));
    __builtin_nontemporal_store(((v4f*)&acc0)[1], (v4f*)(o0 + 4));
    __builtin_nontemporal_store(((v4f*)&acc1)[0], (v4f*)(o1));
    __builtin_nontemporal_store(((v4f*)&acc1)[1], (v4f*)(o1 + 4));

    sacc0 += acc0;
    sacc1 += acc1;
  }

  // write iter-0 partials: slot = nc*2 + npar
  const int ch = nc * 2 + npar;
  float* o  = spart1 + ((size_t)k * GCH2 + ch) * (Bsz * Dsz);
  float* p0 = o + l15 * Dsz + drow;
  float* p1 = o + (16 + l15) * Dsz + drow;
  *(v4f*)(p0)     = ((v4f*)&sacc0)[0];
  *(v4f*)(p0 + 4) = ((v4f*)&sacc0)[1];
  *(v4f*)(p1)     = ((v4f*)&sacc1)[0];
  *(v4f*)(p1 + 4) = ((v4f*)&sacc1)[1];
}

// ---------------------------------------------------------------------------
// Routing iter 0: reduce GEMM partials, c uniform (1/32), squash -> v1.
// One block per (b,k).
// ---------------------------------------------------------------------------
__global__ __launch_bounds__(256) void route_first(const float* __restrict__ spart1,
                                                   float* __restrict__ vout) {
  __shared__ float red[256];
  __shared__ float sd[Dsz];
  const int b = blockIdx.x >> 5;
  const int k = blockIdx.x & 31;
  const int t = threadIdx.x;
  const int d = t & 63;
  const int g = t >> 6;                 // 4 chunk-groups

  const float* base = spart1 + ((size_t)k * GCH2 * Bsz + b) * Dsz + d;
  float acc = 0.f;
  for (int c = g; c < GCH2; c += 4)
    acc += base[(size_t)c * (Bsz * Dsz)];
  red[t] = acc;
  __syncthreads();

  if (g == 0) {
    float s = (red[d] + red[64 + d] + red[128 + d] + red[192 + d]) * (1.0f / 32.0f);
    sd[d] = s;
  }
  __syncthreads();
  if (g == 0) red[d] = sd[d] * sd[d];
  __syncthreads();
  for (int off = 32; off > 0; off >>= 1) {
    if (t < off) red[t] += red[t + off];
    __syncthreads();
  }
  if (g == 0) {
    const float sq = red[0];
    const float sc = (sq / (1.0f + sq)) / sqrtf(sq + 1e-7f);
    vout[((size_t)b * Ksz + k) * Dsz + d] = sd[d] * sc;
  }
}

// ---------------------------------------------------------------------------
// Routing iters 1/2 streaming pass. Block = (b, n-chunk). For each n:
//   t[k] = b_prev[b,k,n] + dot_d(v[b,k,:], uh[k,n,b,:])
//   c[k] = softmax_k(t)            (softmax is over k only -> local!)
//   s_part[k,d] += c[k]*uh[k,n,b,d]
// Optionally stores t[k] as the new logits (iter 1 -> b2 for iter 2).
// ---------------------------------------------------------------------------
__global__ __launch_bounds__(256) void route_pass(const float* __restrict__ uh,
                                                  const float* __restrict__ vin,
                                                  const float* __restrict__ bin,
                                                  float* __restrict__ bout,
                                                  float* __restrict__ spart) {
  __shared__ float vsh[Ksz * Dsz];   // 8 KB
  __shared__ float sp[Ksz * Dsz];    // 8 KB partial s
  __shared__ float tl[Ksz];

  const int b  = blockIdx.x >> 5;          // 32 batches
  const int nc = blockIdx.x & (NCHUNK - 1);
  const int t  = threadIdx.x;
  const int k  = t >> 3;
  const int l8 = t & 7;
  const int d0 = l8 * 8;

  { // load v[b] into LDS
    const float* src = vin + ((size_t)b * Ksz + k) * Dsz + d0;
    *(v4f*)(vsh + k * Dsz + d0)     = *(const v4f*)(src);
    *(v4f*)(vsh + k * Dsz + d0 + 4) = *(const v4f*)(src + 4);
  }
  for (int j = t; j < Ksz * Dsz; j += 256) sp[j] = 0.f;
  __syncthreads();

  for (int nn = 0; nn < NPERCHUNK; ++nn) {
    const int n = nc * NPERCHUNK + nn;
    const float* up = uh + (((size_t)k * Nsz + n) * Bsz + b) * Dsz + d0;
    v4f u0 = __builtin_nontemporal_load((const v4f*)(up));
    v4f u1 = __builtin_nontemporal_load((const v4f*)(up + 4));
    const float* vp = vsh + k * Dsz + d0;

    float p = u0.x * vp[0] + u0.y * vp[1] + u0.z * vp[2] + u0.w * vp[3]
            + u1.x * vp[4] + u1.y * vp[5] + u1.z * vp[6] + u1.w * vp[7];
    p += __shfl_down(p, 4, 8);
    p += __shfl_down(p, 2, 8);
    p += __shfl_down(p, 1, 8);
    if (l8 == 0) {
      float tv = p;
      if (bin)  tv += bin[((size_t)b * Ksz + k) * Nsz + n];
      tl[k] = tv;
      if (bout) bout[((size_t)b * Ksz + k) * Nsz + n] = tv;
    }
    __syncthreads();

    // softmax over the 32 k's (redundant per-thread reduction: cheap)
    float m = -1e30f;
#pragma unroll
    for (int j = 0; j < Ksz; ++j) m = fmaxf(m, tl[j]);
    float sum = 0.f;
#pragma unroll
    for (int j = 0; j < Ksz; ++j) sum += __expf(tl[j] - m);
    const float c = __expf(tl[k] - m) / sum;

    // accumulate (each thread owns unique (k, d-slice))
    float* spp = sp + k * Dsz + d0;
    spp[0] += c * u0.x; spp[1] += c * u0.y; spp[2] += c * u0.z; spp[3] += c * u0.w;
    spp[4] += c * u1.x; spp[5] += c * u1.y; spp[6] += c * u1.z; spp[7] += c * u1.w;
    __syncthreads();
  }

  // write partials: spart[b][nc][k][d]
  float* o = spart + (((size_t)b * NCHUNK + nc) * Ksz + k) * Dsz + d0;
  *(v4f*)(o)     = *(const v4f*)(sp + k * Dsz + d0);
  *(v4f*)(o + 4) = *(const v4f*)(sp + k * Dsz + d0 + 4);
}

// ---------------------------------------------------------------------------
// Reduce chunk partials over n-chunks and apply squash. Block = (b,k), 64 thr.
// ---------------------------------------------------------------------------
__global__ __launch_bounds__(64) void reduce_squash(const float* __restrict__ spart,
                                                    float* __restrict__ vout) {
  __shared__ float red[Dsz];
  const int b = blockIdx.x >> 5;
  const int k = blockIdx.x & 31;
  const int d = threadIdx.x;

  float s = 0.f;
  for (int c = 0; c < NCHUNK; ++c)
    s += spart[(((size_t)b * NCHUNK + c) * Ksz + k) * Dsz + d];

  red[d] = s * s;
  __syncthreads();
  for (int off = 32; off > 0; off >>= 1) {
    if (d < off) red[d] += red[d + off];
    __syncthreads();
  }
  const float sq = red[0];
  const float sc = (sq / (1.0f + sq)) / sqrtf(sq + 1e-7f);
  vout[((size_t)b * Ksz + k) * Dsz + d] = s * sc;
}

// ---------------------------------------------------------------------------
extern "C" void kernel_launch(void* const* d_in, const int* in_sizes, int n_in,
                              void* d_out, int out_size, void* d_ws, size_t ws_size,
                              hipStream_t stream) {
  (void)in_sizes; (void)n_in; (void)out_size; (void)ws_size;
  const float* x = (const float*)d_in[0];   // (32, 2048, 64)
  const float* W = (const float*)d_in[1];   // (32, 2048, 64, 64)
  float* out = (float*)d_out;               // (32, 32, 64)

  float* ws     = (float*)d_ws;
  float* uh     = ws;                                      // 134,217,728 floats
  float* v1     = uh + (size_t)Ksz * Nsz * Bsz * Dsz;      // 65,536
  float* v2     = v1 + (size_t)Bsz * Ksz * Dsz;            // 65,536
  float* b2     = v2 + (size_t)Bsz * Ksz * Dsz;            // 2,097,152
  float* spart  = b2 + (size_t)Bsz * Ksz * Nsz;            // 2,097,152
  float* spart1 = spart + (size_t)Bsz * NCHUNK * Ksz * Dsz;// 8,388,608

  // Phase 1: u_hat GEMM (WMMA f32) + fused iter-0 partial sums
  uhat_gemm<<<Ksz * GCHUNK, 256, 0, stream>>>(x, W, uh, spart1);
  // Iter 0: reduce partials, uniform coupling -> v1
  route_first<<<Bsz * Ksz, 256, 0, stream>>>(spart1, v1);
  // Iter 1: logits = dot(v1, uh); store logits (b2); accumulate -> v2
  route_pass<<<Bsz * NCHUNK, 256, 0, stream>>>(uh, v1, nullptr, b2, spart);
  reduce_squash<<<Bsz * Ksz, 64, 0, stream>>>(spart, v2);
  // Iter 2: logits = b2 + dot(v2, uh); final outputs
  route_pass<<<Bsz * NCHUNK, 256, 0, stream>>>(uh, v2, b2, nullptr, spart);
  reduce_squash<<<Bsz * Ksz, 64, 0, stream>>>(spart, out);
}